// WavBEST_44392781971511
// MI455X (gfx1250) — compile-verified
//
#include <hip/hip_runtime.h>

typedef float v2f __attribute__((ext_vector_type(2)));
typedef float v8f __attribute__((ext_vector_type(8)));

#define CC 64
#define DD 8
#define HH 64
#define WW 64
#define APAD 68                     // padded cin stride (4-bank lane step, conflict-free)
#define AROW 66                     // w positions incl. halo

__device__ __forceinline__ float swish_f(float v) {
    return v / (1.0f + __expf(-v));
}

// Kernel A: e[b,c] = embed . dense1_w[c] + b1[c];  s[b,c] = prompt . dense2_w[c] + b2[c]
__global__ __launch_bounds__(256) void dense_es_kernel(
    const float* __restrict__ embed, const float* __restrict__ prompt,
    const float* __restrict__ d1w, const float* __restrict__ d1b,
    const float* __restrict__ d2w, const float* __restrict__ d2b,
    float* __restrict__ es)   // es[0:256]=e, es[256:512]=s
{
    const int t = threadIdx.x;
    const int bb = t >> 6, c = t & 63;
    const float* eb = embed + bb * 256;
    const float* pb = prompt + bb * 256;
    const float* w1 = d1w + c * 256;
    const float* w2 = d2w + c * 256;
    float a1 = 0.f, a2 = 0.f;
    for (int k = 0; k < 256; ++k) {
        a1 = fmaf(eb[k], w1[k], a1);
        a2 = fmaf(pb[k], w2[k], a2);
    }
    es[t]       = a1 + d1b[c];
    es[256 + t] = a2 + d2b[c];
}

// Implicit-GEMM 3x3x3 conv. One workgroup per (b, d, row-pair): M = 2 rows x 64 w = 128,
// N = 64 out-channels. 8 waves x (2 M-tiles x 2 N-tiles), K = cin*27 stepped by 4.
// LDS layouts put the K-pair contiguous so fragments are single ds_load_b64:
//   Alds[(r*AROW + wpos)*APAD + cin]   (activations, transposed)
//   Wlds[(kw*CC + cout)*APAD + cin]    (weights, transposed)
// MODE 0: in = swish(x + e[b,c]); out = swish(conv + bias) * s[b,o]  -> hbuf
// MODE 1: in = hbuf;              out = conv + x (residual)          -> dst
template<int MODE>
__global__ __launch_bounds__(256) void conv3x3x3_wmma_kernel(
    const float* __restrict__ src,
    const float* __restrict__ wgt,    // (O=64, I=64, 3,3,3)
    const float* __restrict__ bias,   // MODE 0 only
    const float* __restrict__ es,     // MODE 0 only: [e(256) | s(256)]
    const float* __restrict__ resid,  // MODE 1 only: x
    float* __restrict__ dst)
{
    __shared__ float Alds[2 * AROW * APAD];   // 35,904 B
    __shared__ float Wlds[3 * CC * APAD];     // 52,224 B

    const int tid    = threadIdx.x;
    const int lane   = tid & 31;
    const int wv     = tid >> 5;
    const int lane16 = lane & 15;
    const int half   = lane >> 4;     // 0: K={0,1}, 1: K={2,3} within a K4 chunk
    const int koff   = half * 2;
    const int mpair  = wv & 3;        // w-block 0..3 (16 outputs each, both rows)
    const int npair  = wv >> 2;       // out-channel group: {npair*32, npair*32+16}

    const int wg = blockIdx.x;
    const int hb = wg & 31;           // row pair
    const int d  = (wg >> 5) & 7;
    const int b  = wg >> 8;
    const int h0 = hb * 2;

    v8f acc00 = {}, acc01 = {}, acc10 = {}, acc11 = {};

    for (int kd = 0; kd < 3; ++kd) {
        const int zd = d + kd - 1;
        if (zd < 0 || zd >= DD) continue;            // uniform across block
        for (int kh = 0; kh < 3; ++kh) {
            __syncthreads();
            // Stage 2 activation rows (zh = h0+kh-1+r), zero OOB, transposed [r][w][c]
            for (int i = tid; i < 2 * CC * AROW; i += 256) {
                const int r    = i / (CC * AROW);
                const int j    = i - r * (CC * AROW);
                const int c    = j / AROW;
                const int wpos = j - c * AROW;
                const int wi   = wpos - 1;
                const int zh   = h0 + kh - 1 + r;
                float v = 0.f;
                if (wi >= 0 && wi < WW && zh >= 0 && zh < HH) {
                    v = src[(((size_t)(b * CC + c) * DD + zd) * HH + zh) * WW + wi];
                    if (MODE == 0) v = swish_f(v + es[b * CC + c]);
                }
                Alds[(r * AROW + wpos) * APAD + c] = v;
            }
            // Stage weight slab for (kd,kh), transposed [kw][cout][cin]
            for (int i = tid; i < 3 * CC * CC; i += 256) {
                const int c  = i & 63;
                const int o  = (i >> 6) & 63;
                const int kw = i >> 12;
                Wlds[(kw * CC + o) * APAD + c] =
                    wgt[(size_t)(o * CC + c) * 27 + (kd * 9 + kh * 3 + kw)];
            }
            __syncthreads();
            // K-loop: 3 kw taps x 16 chunks of K=4 over cin; 4 b64 loads feed 4 WMMAs
            for (int kw = 0; kw < 3; ++kw) {
                const int aw = mpair * 16 + lane16 + kw;   // halo-shifted w index
                const float* ap0 = &Alds[aw * APAD];
                const float* ap1 = &Alds[(AROW + aw) * APAD];
                const float* bp0 = &Wlds[(kw * CC + npair * 32 + lane16) * APAD];
                const float* bp1 = bp0 + 16 * APAD;
                #pragma unroll
                for (int ck = 0; ck < 16; ++ck) {
                    const int c0 = ck * 4 + koff;
                    const v2f a0 = *(const v2f*)(ap0 + c0);
                    const v2f a1 = *(const v2f*)(ap1 + c0);
                    const v2f b0 = *(const v2f*)(bp0 + c0);
                    const v2f b1 = *(const v2f*)(bp1 + c0);
                    acc00 = __builtin_amdgcn_wmma_f32_16x16x4_f32(
                        false, a0, false, b0, (short)0, acc00, false, false);
                    acc01 = __builtin_amdgcn_wmma_f32_16x16x4_f32(
                        false, a0, false, b1, (short)0, acc01, false, false);
                    acc10 = __builtin_amdgcn_wmma_f32_16x16x4_f32(
                        false, a1, false, b0, (short)0, acc10, false, false);
                    acc11 = __builtin_amdgcn_wmma_f32_16x16x4_f32(
                        false, a1, false, b1, (short)0, acc11, false, false);
                }
            }
        }
    }

    // Epilogue: lane holds 8 contiguous w outputs per tile -> two b128 stores each
    const int wbase = mpair * 16 + half * 8;
    #pragma unroll
    for (int r = 0; r < 2; ++r) {
        #pragma unroll
        for (int t = 0; t < 2; ++t) {
            const v8f acc = r ? (t ? acc11 : acc10) : (t ? acc01 : acc00);
            const int o = npair * 32 + t * 16 + lane16;
            const size_t off =
                (((size_t)(b * CC + o) * DD + d) * HH + (h0 + r)) * WW + wbase;
            float v[8];
            #pragma unroll
            for (int i = 0; i < 8; ++i) v[i] = acc[i];
            if (MODE == 0) {
                const float bo = bias[o];
                const float so = es[256 + b * CC + o];
                #pragma unroll
                for (int i = 0; i < 8; ++i) v[i] = swish_f(v[i] + bo) * so;
            } else {
                const float4 x0 = *(const float4*)(resid + off);
                const float4 x1 = *(const float4*)(resid + off + 4);
                v[0] += x0.x; v[1] += x0.y; v[2] += x0.z; v[3] += x0.w;
                v[4] += x1.x; v[5] += x1.y; v[6] += x1.z; v[7] += x1.w;
            }
            *(float4*)(dst + off)     = make_float4(v[0], v[1], v[2], v[3]);
            *(float4*)(dst + off + 4) = make_float4(v[4], v[5], v[6], v[7]);
        }
    }
}

extern "C" void kernel_launch(void* const* d_in, const int* in_sizes, int n_in,
                              void* d_out, int out_size, void* d_ws, size_t ws_size,
                              hipStream_t stream) {
    (void)in_sizes; (void)n_in; (void)out_size; (void)ws_size;
    const float* x      = (const float*)d_in[0];
    const float* embed  = (const float*)d_in[1];
    const float* prompt = (const float*)d_in[2];
    const float* c20w   = (const float*)d_in[3];
    const float* c20b   = (const float*)d_in[4];
    const float* c21w   = (const float*)d_in[5];
    const float* d1w    = (const float*)d_in[6];
    const float* d1b    = (const float*)d_in[7];
    const float* d2w    = (const float*)d_in[8];
    const float* d2b    = (const float*)d_in[9];
    float* out  = (float*)d_out;
    float* es   = (float*)d_ws;        // 512 floats: e then s
    float* hbuf = es + 512;            // B*C*D*H*W floats (fits in ws)

    dense_es_kernel<<<1, 256, 0, stream>>>(embed, prompt, d1w, d1b, d2w, d2b, es);

    const int ngrid = 4 * DD * (HH / 2);   // one WG per (b, d, row-pair)
    conv3x3x3_wmma_kernel<0><<<ngrid, 256, 0, stream>>>(x, c20w, c20b, es, nullptr, hbuf);
    conv3x3x3_wmma_kernel<1><<<ngrid, 256, 0, stream>>>(hbuf, c21w, nullptr, nullptr, x, out);
}